// HWTV_20409684590797
// MI455X (gfx1250) — compile-verified
//
#include <hip/hip_runtime.h>
#include <hip/hip_bf16.h>
#include <math.h>

// ---------------------------------------------------------------------------
// ADMM TV solver, 20 iterations, 124 independent 128x128 images.
// 2D FFTs implemented as 128x128 DFT matmuls on the CDNA5 matrix engine
// (V_WMMA_F32_16X16X4_F32).  Everything stays L2-resident (~122 MB << 192 MB).
// GEMM passes: grid = (image, tile-row); 8 waves per block, one 16x16 output
// tile per wave -> 7936 waves machine-wide for latency hiding; A-side
// fragments are identical across the block's waves (L0-served).
// ---------------------------------------------------------------------------

typedef __attribute__((ext_vector_type(2))) float v2f;
typedef __attribute__((ext_vector_type(8))) float v8f;

#define NIMG 124
#define IMG_ELEMS 16384            // 128*128
#define NE (NIMG * IMG_ELEMS)      // 2,031,616
#define LD 128
#define ITE 20

// ----------------------------- WMMA helpers --------------------------------

static __device__ __forceinline__ v8f wmma_f32(v2f a, v2f b, v8f c) {
  // D(16x16) = A(16x4) * B(4x16) + C ; f32 everywhere.
  return __builtin_amdgcn_wmma_f32_16x16x4_f32(false, a, false, b,
                                               (short)0, c, false, false);
}

// A-matrix 16x4 f32 layout: VGPR0: K=0 (lanes 0-15), K=2 (lanes 16-31);
//                           VGPR1: K=1 (lanes 0-15), K=3 (lanes 16-31).
static __device__ __forceinline__ v2f afrag(const float* __restrict__ M,
                                            int mBase, int k, int r, int half) {
  const float* p = M + (size_t)(mBase + r) * LD + (k + 2 * half);
  v2f a;
  a.x = p[0];
  a.y = p[1];
  return a;
}

// B-matrix 4x16 f32 layout (rows striped across lanes within a VGPR):
// VGPR0: K=0 (lanes 0-15), K=2 (lanes 16-31); VGPR1: K=1, K=3.  N = lane&15.
static __device__ __forceinline__ v2f bfrag(const float* __restrict__ M,
                                            int k, int nBase, int r, int half) {
  int row = k + 2 * half;
  v2f b;
  b.x = M[(size_t)row * LD + nBase + r];
  b.y = M[(size_t)(row + 1) * LD + nBase + r];
  return b;
}

// C/D 16x16 f32 layout: VGPR j holds M=j (lanes 0-15) and M=j+8 (lanes 16-31).
static __device__ __forceinline__ void store_tile(float* __restrict__ M,
                                                  int mBase, int nBase, int r,
                                                  int half, v8f c, float s) {
#pragma unroll
  for (int j = 0; j < 8; ++j)
    M[(size_t)(mBase + j + 8 * half) * LD + nBase + r] = c[j] * s;
}

// ----------------------------- small kernels -------------------------------

// DFT tables: F = C + iS with C[m][k]=cos(2pi mk/128), S[m][k]=-sin(2pi mk/128),
// NS = -S = +sin (used to realize all subtractions as accumulations, since
// f32 WMMA NEG bits only support C-negate, not A/B-negate).
__global__ void k_tables(float* Ct, float* St, float* NSt) {
  int i = blockIdx.x * blockDim.x + threadIdx.x;
  if (i >= IMG_ELEMS) return;
  int m = i >> 7, k = i & 127;
  int p = (m * k) & 127;  // exact mod-128 phase reduction
  float ang = (6.283185307179586f / 128.f) * (float)p;
  float s, c;
  sincosf(ang, &s, &c);
  Ct[i] = c;
  St[i] = -s;
  NSt[i] = s;
}

__global__ void k_init(const float* __restrict__ Y, float* Z, float* X,
                       float* G1, float* G21, float* G22, float* bufH,
                       float* bufV) {
  int idx = blockIdx.x * blockDim.x + threadIdx.x;
  int w = idx & 127, h = (idx >> 7) & 127;
  int ib = idx & ~(IMG_ELEMS - 1);
  float y = Y[idx];
  Z[idx] = y;
  X[idx] = y;
  G1[idx] = 0.f;
  G21[idx] = 0.f;
  G22[idx] = 0.f;
  bufH[idx] = y - Y[ib + (h << 7) + ((w + 127) & 127)];
  bufV[idx] = y - Y[ib + (((h + 127) & 127) << 7) + w];
}

static __device__ __forceinline__ float shrinkf(float x, float t) {
  float m = fmaxf(fabsf(x) - t, 0.f);
  return copysignf(m, x);
}

// Uh/Uv shrink + build GEMM right-hand sides:
//   bufC <- A = X + G1/mu1 ; bufH <- B = Uh + G21/mu2 ; bufV <- C = Uv + G22/mu2
__global__ void k_pre(const float* __restrict__ X, const float* __restrict__ G1,
                      const float* __restrict__ G21,
                      const float* __restrict__ G22, float* bufH, float* bufV,
                      float* bufC, float im1, float im2, float thr) {
  int idx = blockIdx.x * blockDim.x + threadIdx.x;
  float g21 = G21[idx] * im2;
  float g22 = G22[idx] * im2;
  float uh = shrinkf(bufH[idx] - g21, thr);  // bufH held DhZ
  float uv = shrinkf(bufV[idx] - g22, thr);  // bufV held DvZ
  bufC[idx] = X[idx] + G1[idx] * im1;
  bufH[idx] = uh + g21;
  bufV[idx] = uv + g22;
}

// Pointwise Fourier filter.  MBC packs fft2(B + iC); Hermitian split recovers
// FB, FC.  Writes NUM = (mu1 FA + mu2 (FHc FB + FVc FC)) / (mu1 + mu2 demo).
__global__ void k_filter(const float* __restrict__ MAr,
                         const float* __restrict__ MAi,
                         const float* __restrict__ MBr,
                         const float* __restrict__ MBi, float* Nr, float* Ni,
                         float mu1, float mu2) {
  int idx = blockIdx.x * blockDim.x + threadIdx.x;
  int v = idx & 127, u = (idx >> 7) & 127;
  int ib = idx & ~(IMG_ELEMS - 1);
  int um = (128 - u) & 127, vm = (128 - v) & 127;
  int midx = ib + (um << 7) + vm;
  float far_ = MAr[idx], fai = MAi[idx];
  float pr = MBr[idx], pi = MBi[idx];
  float qr = MBr[midx], qi = MBi[midx];
  // FB = (P + conj(Q))/2 ; FC = (P - conj(Q))/(2i)
  float fbr = 0.5f * (pr + qr), fbi = 0.5f * (pi - qi);
  float fcr = 0.5f * (pi + qi), fci = -0.5f * (pr - qr);
  const float c0 = 6.283185307179586f / 128.f;
  float su, cu, sv, cv;
  sincosf(c0 * (float)u, &su, &cu);
  sincosf(c0 * (float)v, &sv, &cv);
  // FH = 1 - e^{-i th_v}  =>  FHc = (1-cos v) - i sin v ; FVc analogous on u.
  float hr = 1.f - cv, hi = -sv;
  float vr = 1.f - cu, vi = -su;
  float t1r = hr * fbr - hi * fbi, t1i = hr * fbi + hi * fbr;
  float t2r = vr * fcr - vi * fci, t2i = vr * fci + vi * fcr;
  float nr = mu1 * far_ + mu2 * (t1r + t2r);
  float ni = mu1 * fai + mu2 * (t1i + t2i);
  float den = mu1 + mu2 * (4.f - 2.f * cv - 2.f * cu);
  Nr[idx] = nr / den;
  Ni[idx] = ni / den;
}

// Post-solve elementwise updates (diffs of new Z, X/G updates).
// bufH holds B = Uh + G21/mu2 -> Uh = bufH - G21/mu2 (G21 still old). Same bufV.
__global__ void k_post(const float* __restrict__ Z, const float* __restrict__ Y,
                       const float* __restrict__ inW, float* X, float* G1,
                       float* G21, float* G22, float* bufH, float* bufV,
                       float mu1, float mu2, float im2) {
  int idx = blockIdx.x * blockDim.x + threadIdx.x;
  int w = idx & 127, h = (idx >> 7) & 127;
  int ib = idx & ~(IMG_ELEMS - 1);
  float z = Z[idx];
  float dh = z - Z[ib + (h << 7) + ((w + 127) & 127)];
  float dv = z - Z[ib + (((h + 127) & 127) << 7) + w];
  float g21 = G21[idx], g22 = G22[idx];
  float uh = bufH[idx] - g21 * im2;
  float uv = bufV[idx] - g22 * im2;
  float iw = inW[idx];
  float x = (iw * Y[idx] + mu1 * z - G1[idx]) / (iw + mu1);
  X[idx] = x;
  G1[idx] += mu1 * (x - z);
  G21[idx] = g21 + mu2 * (uh - dh);
  G22[idx] = g22 + mu2 * (uv - dv);
  bufH[idx] = dh;
  bufV[idx] = dv;
}

// ----------------------------- GEMM passes ---------------------------------
// grid = (image, tile-row).  Block = 256 threads = 8 waves; wave w computes
// output tile (blockIdx.y, w).  A-side fragments are identical across the
// block's 8 waves -> WGP$ broadcast; B-side fragments differ per wave.

// Pass1: T = F * [A, B+iC]   (left = tables, right = data)
__global__ void pass1(const float* __restrict__ Ct, const float* __restrict__ St,
                      const float* __restrict__ NSt,
                      const float* __restrict__ RA, const float* __restrict__ RB,
                      const float* __restrict__ RC, float* TAr, float* TAi,
                      float* TBr, float* TBi) {
  size_t base = (size_t)blockIdx.x * IMG_ELEMS;
  const float* ra = RA + base;
  const float* rb = RB + base;
  const float* rc = RC + base;
  int wave = threadIdx.x >> 5, lane = threadIdx.x & 31;
  int r = lane & 15, half = lane >> 4;
  int mB = blockIdx.y << 4;
  int nB = wave << 4;
  v8f ar = {}, ai = {}, br = {}, bi = {};
  for (int k = 0; k < 128; k += 4) {
    v2f aC = afrag(Ct, mB, k, r, half);
    v2f aS = afrag(St, mB, k, r, half);
    v2f aN = afrag(NSt, mB, k, r, half);
    v2f bA = bfrag(ra, k, nB, r, half);
    v2f bB = bfrag(rb, k, nB, r, half);
    v2f bCc = bfrag(rc, k, nB, r, half);
    ar = wmma_f32(aC, bA, ar);                      // TAr = C*A
    ai = wmma_f32(aS, bA, ai);                      // TAi = S*A
    br = wmma_f32(aC, bB, br);
    br = wmma_f32(aN, bCc, br);                     // TBr = C*B - S*C
    bi = wmma_f32(aS, bB, bi);
    bi = wmma_f32(aC, bCc, bi);                     // TBi = S*B + C*C
  }
  store_tile(TAr + base, mB, nB, r, half, ar, 1.f);
  store_tile(TAi + base, mB, nB, r, half, ai, 1.f);
  store_tile(TBr + base, mB, nB, r, half, br, 1.f);
  store_tile(TBi + base, mB, nB, r, half, bi, 1.f);
}

// Pass2: M = T * F   (left = data, right = tables)
__global__ void pass2(const float* __restrict__ Ct, const float* __restrict__ St,
                      const float* __restrict__ NSt,
                      const float* __restrict__ TAr,
                      const float* __restrict__ TAi,
                      const float* __restrict__ TBr,
                      const float* __restrict__ TBi, float* MAr, float* MAi,
                      float* MBr, float* MBi) {
  size_t base = (size_t)blockIdx.x * IMG_ELEMS;
  const float* tar = TAr + base;
  const float* tai = TAi + base;
  const float* tbr = TBr + base;
  const float* tbi = TBi + base;
  int wave = threadIdx.x >> 5, lane = threadIdx.x & 31;
  int r = lane & 15, half = lane >> 4;
  int mB = blockIdx.y << 4;
  int nB = wave << 4;
  v8f ar = {}, ai = {}, br = {}, bi = {};
  for (int k = 0; k < 128; k += 4) {
    v2f aAr = afrag(tar, mB, k, r, half);
    v2f aAi = afrag(tai, mB, k, r, half);
    v2f aBr = afrag(tbr, mB, k, r, half);
    v2f aBi = afrag(tbi, mB, k, r, half);
    v2f bC = bfrag(Ct, k, nB, r, half);
    v2f bS = bfrag(St, k, nB, r, half);
    v2f bN = bfrag(NSt, k, nB, r, half);
    ar = wmma_f32(aAr, bC, ar);
    ar = wmma_f32(aAi, bN, ar);                     // MAr = TAr*C - TAi*S
    ai = wmma_f32(aAr, bS, ai);
    ai = wmma_f32(aAi, bC, ai);                     // MAi = TAr*S + TAi*C
    br = wmma_f32(aBr, bC, br);
    br = wmma_f32(aBi, bN, br);
    bi = wmma_f32(aBr, bS, bi);
    bi = wmma_f32(aBi, bC, bi);
  }
  store_tile(MAr + base, mB, nB, r, half, ar, 1.f);
  store_tile(MAi + base, mB, nB, r, half, ai, 1.f);
  store_tile(MBr + base, mB, nB, r, half, br, 1.f);
  store_tile(MBi + base, mB, nB, r, half, bi, 1.f);
}

// Pass3: U = conj(F) * NUM   (left = tables)
__global__ void pass3(const float* __restrict__ Ct, const float* __restrict__ St,
                      const float* __restrict__ NSt,
                      const float* __restrict__ Nr, const float* __restrict__ Ni,
                      float* Ur, float* Ui) {
  size_t base = (size_t)blockIdx.x * IMG_ELEMS;
  const float* nr = Nr + base;
  const float* ni = Ni + base;
  int wave = threadIdx.x >> 5, lane = threadIdx.x & 31;
  int r = lane & 15, half = lane >> 4;
  int mB = blockIdx.y << 4;
  int nB = wave << 4;
  v8f ur = {}, ui = {};
  for (int k = 0; k < 128; k += 4) {
    v2f aC = afrag(Ct, mB, k, r, half);
    v2f aS = afrag(St, mB, k, r, half);
    v2f aN = afrag(NSt, mB, k, r, half);
    v2f bR = bfrag(nr, k, nB, r, half);
    v2f bI = bfrag(ni, k, nB, r, half);
    ur = wmma_f32(aC, bR, ur);
    ur = wmma_f32(aS, bI, ur);                      // Ur = C*Nr + S*Ni
    ui = wmma_f32(aC, bI, ui);
    ui = wmma_f32(aN, bR, ui);                      // Ui = C*Ni - S*Nr
  }
  store_tile(Ur + base, mB, nB, r, half, ur, 1.f);
  store_tile(Ui + base, mB, nB, r, half, ui, 1.f);
}

// Pass4: Z = real(U * conj(F)) / 16384   (right = tables)
__global__ void pass4(const float* __restrict__ Ct, const float* __restrict__ St,
                      const float* __restrict__ Ur, const float* __restrict__ Ui,
                      float* Z) {
  size_t base = (size_t)blockIdx.x * IMG_ELEMS;
  const float* ur = Ur + base;
  const float* ui = Ui + base;
  int wave = threadIdx.x >> 5, lane = threadIdx.x & 31;
  int r = lane & 15, half = lane >> 4;
  int mB = blockIdx.y << 4;
  int nB = wave << 4;
  v8f z = {};
  for (int k = 0; k < 128; k += 4) {
    v2f aR = afrag(ur, mB, k, r, half);
    v2f aI = afrag(ui, mB, k, r, half);
    v2f bC = bfrag(Ct, k, nB, r, half);
    v2f bS = bfrag(St, k, nB, r, half);
    z = wmma_f32(aR, bC, z);
    z = wmma_f32(aI, bS, z);                        // Re(U*F*) = Ur*C + Ui*S
  }
  store_tile(Z + base, mB, nB, r, half, z, 1.f / 16384.f);
}

// ----------------------------- host driver ---------------------------------

extern "C" void kernel_launch(void* const* d_in, const int* in_sizes, int n_in,
                              void* d_out, int out_size, void* d_ws,
                              size_t ws_size, hipStream_t stream) {
  (void)in_sizes; (void)n_in; (void)out_size; (void)ws_size;
  const float* Y = (const float*)d_in[0];
  const float* inW = (const float*)d_in[1];
  float* Z = (float*)d_out;  // Z lives in d_out across all iterations
  float* ws = (float*)d_ws;

  // workspace layout (floats): 4 table slots (pad) + 15 image-set buffers
  float* Ct = ws;
  float* St = Ct + IMG_ELEMS;
  float* NSt = St + IMG_ELEMS;
  float* p = ws + 4 * (size_t)IMG_ELEMS;
  float* X = p;            p += NE;
  float* G1 = p;           p += NE;
  float* G21 = p;          p += NE;
  float* G22 = p;          p += NE;
  float* bufH = p;         p += NE;   // DhZ -> then B-RHS
  float* bufV = p;         p += NE;   // DvZ -> then C-RHS
  float* bufC = p;         p += NE;   // A-RHS
  float* TAr = p;          p += NE;   // also NUMr
  float* TAi = p;          p += NE;   // also NUMi
  float* TBr = p;          p += NE;   // also Ur
  float* TBi = p;          p += NE;   // also Ui
  float* MAr = p;          p += NE;
  float* MAi = p;          p += NE;
  float* MBr = p;          p += NE;
  float* MBi = p;          p += NE;   // total ~122 MB (L2-resident)

  const int EB = 256, EG = NE / EB;   // 7936 blocks, exact
  const dim3 gg(NIMG, 8);             // image x tile-row -> 992 blocks, 8 waves

  k_tables<<<IMG_ELEMS / 256, 256, 0, stream>>>(Ct, St, NSt);
  k_init<<<EG, EB, 0, stream>>>(Y, Z, X, G1, G21, G22, bufH, bufV);

  float m1 = 0.1f, m2 = 0.1f;  // MU1, MU2; RHO=1.05 in f32 like the reference
  for (int it = 0; it < ITE; ++it) {
    float im1 = 1.f / m1, im2 = 1.f / m2, thr = 0.1f / m2;  // LAM = 0.1
    k_pre<<<EG, EB, 0, stream>>>(X, G1, G21, G22, bufH, bufV, bufC, im1, im2,
                                 thr);
    pass1<<<gg, 256, 0, stream>>>(Ct, St, NSt, bufC, bufH, bufV, TAr, TAi,
                                  TBr, TBi);
    pass2<<<gg, 256, 0, stream>>>(Ct, St, NSt, TAr, TAi, TBr, TBi, MAr, MAi,
                                  MBr, MBi);
    k_filter<<<EG, EB, 0, stream>>>(MAr, MAi, MBr, MBi, TAr, TAi, m1, m2);
    pass3<<<gg, 256, 0, stream>>>(Ct, St, NSt, TAr, TAi, TBr, TBi);
    pass4<<<gg, 256, 0, stream>>>(Ct, St, TBr, TBi, Z);
    if (it < ITE - 1)
      k_post<<<EG, EB, 0, stream>>>(Z, Y, inW, X, G1, G21, G22, bufH, bufV, m1,
                                    m2, im2);
    m1 *= 1.05f;
    m2 *= 1.05f;
  }
}